// QuantumGenerator_52183852646454
// MI455X (gfx1250) — compile-verified
//
#include <hip/hip_runtime.h>
#include <hip/hip_bf16.h>

// ---------------------------------------------------------------------------
// QuantumGenerator: angles = x @ W^T + b  (via V_WMMA_F32_16X16X4_F32),
// then per-sample 12-qubit state-vector simulation fully in LDS.
// ---------------------------------------------------------------------------

#define NQ      12
#define DIM     4096          // 2^12 amplitudes
#define NPAIR   (DIM / 2)
#define NCPAIR  (DIM / 4)
#define NCIRC   3
#define NANG    216           // 3 * 6 * 12
#define BATCH   128
#define THREADS 256

typedef float v2f __attribute__((ext_vector_type(2)));
typedef float v8f __attribute__((ext_vector_type(8)));

// ------------------------- Kernel 1: angles via WMMA ------------------------
// rot[b][cd] = sum_q x[b,q] * W[cd*12+q] + bias[cd],  cd in [0,216)
// One wave per 16x16 output tile; M = batch (8 tiles), N = cd (14 tiles, padded).
__global__ void __launch_bounds__(32)
angles_gemm_wmma(const float* __restrict__ x, const float* __restrict__ W,
                 const float* __restrict__ bias, float* __restrict__ angles) {
    const int tile = blockIdx.x;
    const int tm = tile & 7;         // 0..7   batch tile
    const int tn = tile >> 3;        // 0..13  cd tile
    const int lane = threadIdx.x;    // wave32
    const int half = lane >> 4;      // 0: K pair {0,1}; 1: K pair {2,3}
    const int l    = lane & 15;

    v8f acc = {};
#pragma unroll
    for (int ks = 0; ks < 12; ks += 4) {
        // A (16x4 f32): lanes 0-15 hold M=l with K=ks,ks+1; lanes 16-31 K=ks+2,ks+3
        const int m  = tm * 16 + l;          // < 128 always
        const int ka = ks + half * 2;        // <= 10, so ka+1 <= 11 < K=12
        v2f a;
        a.x = x[m * NQ + ka];
        a.y = x[m * NQ + ka + 1];
        // B (4x16 f32): lanes 0-15 hold N=l with K=ks (v0), K=ks+1 (v1);
        //               lanes 16-31 K=ks+2 (v0), K=ks+3 (v1)
        const int n = tn * 16 + l;
        v2f bb;
        if (n < NANG) {
            bb.x = W[n * NQ + ka];
            bb.y = W[n * NQ + ka + 1];
        } else {
            bb.x = 0.0f; bb.y = 0.0f;
        }
        acc = __builtin_amdgcn_wmma_f32_16x16x4_f32(
            /*neg_a=*/false, a, /*neg_b=*/false, bb,
            /*c_mod=*/(short)0, acc, /*reuse_a=*/false, /*reuse_b=*/false);
    }
    // C/D layout: VGPR v -> lanes 0-15: M=v, lanes 16-31: M=v+8; N = lane%16
    const int n = tn * 16 + l;
    if (n < NANG) {
        const float bv = bias[n];
#pragma unroll
        for (int v = 0; v < 8; ++v) {
            const int m = tm * 16 + half * 8 + v;
            angles[m * NANG + n] = acc[v] + bv;
        }
    }
}

// ------------------------- Kernel 2: state-vector sim -----------------------
struct Cplx { float r, i; };
__device__ __forceinline__ Cplx cmul(Cplx a, Cplx b) { return {a.r*b.r - a.i*b.i, a.r*b.i + a.i*b.r}; }
__device__ __forceinline__ Cplx cadd(Cplx a, Cplx b) { return {a.r + b.r, a.i + b.i}; }
__device__ __forceinline__ Cplx csub(Cplx a, Cplx b) { return {a.r - b.r, a.i - b.i}; }
__device__ __forceinline__ Cplx cscale(float s, Cplx a) { return {s * a.r, s * a.i}; }

// Apply complex 2x2 U on bit p over all 2048 pairs (cooperative across block).
__device__ __forceinline__ void apply_1q(float* psiR, float* psiI, int p,
                                         Cplx u00, Cplx u01, Cplx u10, Cplx u11,
                                         int tid) {
    const int lowmask = (1 << p) - 1;
#pragma unroll 4
    for (int k = tid; k < NPAIR; k += THREADS) {
        const int i0 = ((k & ~lowmask) << 1) | (k & lowmask);
        const int i1 = i0 | (1 << p);
        const float ar = psiR[i0], ai = psiI[i0];
        const float br = psiR[i1], bi = psiI[i1];
        psiR[i0] = u00.r*ar - u00.i*ai + u01.r*br - u01.i*bi;
        psiI[i0] = u00.r*ai + u00.i*ar + u01.r*bi + u01.i*br;
        psiR[i1] = u10.r*ar - u10.i*ai + u11.r*br - u11.i*bi;
        psiI[i1] = u10.r*ai + u10.i*ar + u11.r*bi + u11.i*br;
    }
    __syncthreads();
}

// Apply controlled-U: control bit pc (must be 1), target bit pt.
__device__ __forceinline__ void apply_c1q(float* psiR, float* psiI, int pc, int pt,
                                          Cplx u00, Cplx u01, Cplx u10, Cplx u11,
                                          int tid) {
    const int pl = pc < pt ? pc : pt;
    const int ph = pc < pt ? pt : pc;
    const int lowmask = (1 << pl) - 1;
    const int midmask = (1 << (ph - 1 - pl)) - 1;
#pragma unroll 4
    for (int k = tid; k < NCPAIR; k += THREADS) {
        const int low  = k & lowmask;
        const int mid  = (k >> pl) & midmask;
        const int high = k >> (ph - 1);
        const int i0 = (high << (ph + 1)) | (mid << (pl + 1)) | low | (1 << pc);
        const int i1 = i0 | (1 << pt);
        const float ar = psiR[i0], ai = psiI[i0];
        const float br = psiR[i1], bi = psiI[i1];
        psiR[i0] = u00.r*ar - u00.i*ai + u01.r*br - u01.i*bi;
        psiI[i0] = u00.r*ai + u00.i*ar + u01.r*bi + u01.i*br;
        psiR[i1] = u10.r*ar - u10.i*ai + u11.r*br - u11.i*bi;
        psiI[i1] = u10.r*ai + u10.i*ar + u11.r*bi + u11.i*br;
    }
    __syncthreads();
}

__global__ void __launch_bounds__(THREADS)
qsim_kernel(const float* __restrict__ angles, float* __restrict__ out) {
    __shared__ float psiR[DIM];
    __shared__ float psiI[DIM];
    __shared__ float ang[NANG];
    __shared__ float red[THREADS];

    const int b   = blockIdx.x;
    const int tid = threadIdx.x;

    for (int i = tid; i < NANG; i += THREADS) ang[i] = angles[b * NANG + i];
    for (int i = tid; i < DIM; i += THREADS) { psiR[i] = 0.0f; psiI[i] = 0.0f; }
    __syncthreads();
    if (tid == 0) psiR[0] = 1.0f;    // |0...0>
    __syncthreads();

    for (int ci = 0; ci < NCIRC; ++ci) {
        const float* wc = &ang[ci * 72];
        // ---- fused single-qubit layer: U = RZ(w3) RY(w2) RZ(w1) RY(w0) ----
        for (int q = 0; q < NQ; ++q) {
            const float w0 = wc[q], w1 = wc[12 + q], w2 = wc[24 + q], w3 = wc[36 + q];
            float c0, s0, c1, s1, c2, s2, c3, s3;
            __sincosf(0.5f * w0, &s0, &c0);
            __sincosf(0.5f * w1, &s1, &c1);
            __sincosf(0.5f * w2, &s2, &c2);
            __sincosf(0.5f * w3, &s3, &c3);
            const Cplx e1  = {c1, -s1}, e1c = {c1, s1};   // e^{-i w1/2}, conj
            const Cplx e3  = {c3, -s3}, e3c = {c3, s3};
            // M1 = RZ(w1) * RY(w0)
            const Cplx m00 = cscale( c0, e1),  m01 = cscale(-s0, e1);
            const Cplx m10 = cscale( s0, e1c), m11 = cscale( c0, e1c);
            // M2 = RY(w2) * M1
            const Cplx t00 = csub(cscale(c2, m00), cscale(s2, m10));
            const Cplx t01 = csub(cscale(c2, m01), cscale(s2, m11));
            const Cplx t10 = cadd(cscale(s2, m00), cscale(c2, m10));
            const Cplx t11 = cadd(cscale(s2, m01), cscale(c2, m11));
            // U = RZ(w3) * M2
            apply_1q(psiR, psiI, 11 - q,
                     cmul(e3, t00), cmul(e3, t01), cmul(e3c, t10), cmul(e3c, t11),
                     tid);
        }
        // ---- fused entangling layer: ctrl-[RZ(w5) RY(w4)] on (q -> q+1) ----
        for (int q = 0; q < NQ; ++q) {
            const int t = (q + 1) % NQ;
            const float w4 = wc[48 + q], w5 = wc[60 + q];
            float c4, s4, c5, s5;
            __sincosf(0.5f * w4, &s4, &c4);
            __sincosf(0.5f * w5, &s5, &c5);
            const Cplx e5 = {c5, -s5}, e5c = {c5, s5};
            const Cplx v00 = cscale( c4, e5),  v01 = cscale(-s4, e5);
            const Cplx v10 = cscale( s4, e5c), v11 = cscale( c4, e5c);
            apply_c1q(psiR, psiI, 11 - q, 11 - t, v00, v01, v10, v11, tid);
        }
    }

    // ---- expectation values: <X_q>, <Z_q> ----
    for (int q = 0; q < NQ; ++q) {
        const int p = 11 - q;
        const int lowmask = (1 << p) - 1;
        float px = 0.0f, pz = 0.0f;
#pragma unroll 4
        for (int k = tid; k < NPAIR; k += THREADS) {
            const int i0 = ((k & ~lowmask) << 1) | (k & lowmask);
            const int i1 = i0 | (1 << p);
            const float ar = psiR[i0], ai = psiI[i0];
            const float br = psiR[i1], bi = psiI[i1];
            px += 2.0f * (ar * br + ai * bi);                 // 2 Re(conj(a) b)
            pz += (ar * ar + ai * ai) - (br * br + bi * bi);  // |a|^2 - |b|^2
        }
        red[tid] = px; __syncthreads();
        for (int s = THREADS / 2; s > 0; s >>= 1) {
            if (tid < s) red[tid] += red[tid + s];
            __syncthreads();
        }
        if (tid == 0) out[b * 24 + q] = red[0];
        __syncthreads();
        red[tid] = pz; __syncthreads();
        for (int s = THREADS / 2; s > 0; s >>= 1) {
            if (tid < s) red[tid] += red[tid + s];
            __syncthreads();
        }
        if (tid == 0) out[b * 24 + 12 + q] = red[0];
        __syncthreads();
    }
}

// ----------------------------------------------------------------------------
extern "C" void kernel_launch(void* const* d_in, const int* in_sizes, int n_in,
                              void* d_out, int out_size, void* d_ws, size_t ws_size,
                              hipStream_t stream) {
    const float* x    = (const float*)d_in[0];   // [128, 12]
    const float* W    = (const float*)d_in[1];   // [3, 72, 12]
    const float* bias = (const float*)d_in[2];   // [3, 72]
    float* out        = (float*)d_out;           // [128, 24]
    float* angles     = (float*)d_ws;            // [128, 216] scratch

    // 8 M-tiles x 14 N-tiles, one wave32 per tile
    angles_gemm_wmma<<<dim3(8 * 14), dim3(32), 0, stream>>>(x, W, bias, angles);
    // one workgroup per batch sample, state vector in LDS
    qsim_kernel<<<dim3(BATCH), dim3(THREADS), 0, stream>>>(angles, out);
}